// KalmanFilter_71124658422067
// MI455X (gfx1250) — compile-verified
//
#include <hip/hip_runtime.h>
#include <hip/hip_bf16.h>
#include <math.h>

// Kalman filter scan: G=2048 chains, T=256 steps, S=32 state, M=8 meas.
// One wave32 per group. P kept in LDS per wave; F/Q WMMA fragments kept in
// registers across the whole T loop. Predict step P' = F P F32 Fᵀ + Q done with
// V_WMMA_F32_16X16X4_F32 (2x2 tiles of 16x16, K=32 as 8 chunks of 4).
// Small M=8 pieces (HP, S_t, Cholesky solve, rank-8 update) on VALU.

#define GG 2048
#define TT 256
#define SS 32
#define MM 8
#define WAVES 4   // waves (=groups) per block -> 128 threads, 512 blocks

typedef float v2f __attribute__((ext_vector_type(2)));
typedef float v8f __attribute__((ext_vector_type(8)));

// Intra-wave LDS producer->consumer fence (lanes are lockstep; only need DS
// completion). Also a compiler reordering barrier.
#define LDS_FENCE() asm volatile("s_wait_dscnt 0" ::: "memory")

static __device__ __forceinline__ v8f wmma4(v2f a, v2f b, v8f c) {
  // (neg_a, A, neg_b, B, c_mod, C, reuse_a, reuse_b)
  return __builtin_amdgcn_wmma_f32_16x16x4_f32(false, a, false, b, (short)0, c,
                                               false, false);
}

__global__ __launch_bounds__(32 * WAVES, 1)
void KalmanFilter_71124658422067_kernel(const float* __restrict__ inp,   // [G,T,M]
                                        const float* __restrict__ F,     // [S,S]
                                        const float* __restrict__ H,     // [M,S]
                                        const float* __restrict__ Q,     // [S,S]
                                        const float* __restrict__ R,     // [M,M]
                                        const float* __restrict__ m0,    // [G,S]
                                        const float* __restrict__ P0,    // [G,S,S]
                                        float* __restrict__ out) {
  // ---- block-shared constants ----
  __shared__ float Fs[SS][SS];
  __shared__ float Hs[MM][SS];
  __shared__ float Qs[SS][SS];
  __shared__ float Rs[MM][MM];
  // ---- per-wave working state ----
  __shared__ float Pw_[WAVES][SS][SS];   // covariance (prior at loop top)
  __shared__ float Uw_[WAVES][SS][SS];   // scratch: P_upd, then T1 = F*Psym
  __shared__ float HPw_[WAVES][MM][SS];
  __shared__ float KTw_[WAVES][MM][SS];
  __shared__ float Scw_[WAVES][MM][MM];  // S_t, Cholesky in place
  __shared__ float mw_[WAVES][SS];
  __shared__ float muw_[WAVES][SS];
  __shared__ float measw_[WAVES][MM];
  __shared__ float rsw_[WAVES][MM];

  const int tid  = threadIdx.x;
  const int wave = tid >> 5;
  const int lane = tid & 31;
  const int g    = blockIdx.x * WAVES + wave;
  const int lhi  = lane >> 4;   // 0/1
  const int llo  = lane & 15;

  // cooperative load of shared constants
  for (int i = tid; i < SS * SS; i += 32 * WAVES) {
    Fs[i >> 5][i & 31] = F[i];
    Qs[i >> 5][i & 31] = Q[i];
  }
  for (int i = tid; i < MM * SS; i += 32 * WAVES) Hs[i >> 5][i & 31] = H[i];
  for (int i = tid; i < MM * MM; i += 32 * WAVES) Rs[i >> 3][i & 7]  = R[i];
  __syncthreads();

  float (*Pw)[SS]  = Pw_[wave];
  float (*Uw)[SS]  = Uw_[wave];
  float (*HPw)[SS] = HPw_[wave];
  float (*KTw)[SS] = KTw_[wave];
  float (*Scw)[MM] = Scw_[wave];
  float* mw   = mw_[wave];
  float* muw  = muw_[wave];
  float* meas = measw_[wave];
  float* rsd  = rsw_[wave];

  // ---- init per-group state ----
  for (int i = 0; i < SS; ++i)
    Pw[i][lane] = P0[(size_t)g * SS * SS + i * SS + lane];
  mw[lane] = m0[(size_t)g * SS + lane];
  LDS_FENCE();

  // ---- register-resident WMMA fragments of F, Fᵀ, Q (constant over T) ----
  // A-frag (16x4): lane row = llo (+16*i), VGPR v: K = 4*k4 + v + 2*lhi
  // B-frag (4x16): lane col = llo (+16*j), VGPR v: K = 4*k4 + 2*v + lhi
  // C-frag (16x16): VGPR v: row = v + 8*lhi, col = llo
  float fa[2][8][2];   // F as A
  float ftb[2][8][2];  // Fᵀ as B  (element (K,N) = F[N][K])
#pragma unroll
  for (int i = 0; i < 2; ++i) {
#pragma unroll
    for (int k4 = 0; k4 < 8; ++k4) {
      fa[i][k4][0]  = Fs[16 * i + llo][4 * k4 + 0 + 2 * lhi];
      fa[i][k4][1]  = Fs[16 * i + llo][4 * k4 + 1 + 2 * lhi];
      ftb[i][k4][0] = Fs[16 * i + llo][4 * k4 + 0 + lhi];
      ftb[i][k4][1] = Fs[16 * i + llo][4 * k4 + 2 + lhi];
    }
  }
  v8f qc[2][2];
#pragma unroll
  for (int i = 0; i < 2; ++i)
#pragma unroll
    for (int j = 0; j < 2; ++j)
#pragma unroll
      for (int v = 0; v < 8; ++v)
        qc[i][j][v] = Qs[16 * i + v + 8 * lhi][16 * j + llo];

  const size_t meanBase = (size_t)g * TT * MM;
  const size_t covBase  = (size_t)GG * TT * MM + (size_t)g * TT * (MM * MM);

  for (int t = 0; t < TT; ++t) {
    // ---- 1. predicted measurement mean of the prior: H @ m  (emit) ----
    if (lane < MM) {
      float acc = 0.f;
      for (int s = 0; s < SS; ++s) acc += Hs[lane][s] * mw[s];
      meas[lane] = acc;
      out[meanBase + (size_t)t * MM + lane] = acc;
    }
    LDS_FENCE();
    if (lane < MM)
      rsd[lane] = inp[((size_t)g * TT + t) * MM + lane] - meas[lane];

    // ---- 2. HP = H @ P  (lane owns column) ----
    for (int r = 0; r < MM; ++r) {
      float acc = 0.f;
      for (int s = 0; s < SS; ++s) acc += Hs[r][s] * Pw[s][lane];
      HPw[r][lane] = acc;
    }
    LDS_FENCE();

    // ---- 3. S_t = HP Hᵀ + R  (emit covs) ----
    {
      const int r0 = lane >> 3, c = lane & 7;
      for (int rr = 0; rr < 2; ++rr) {
        const int r = r0 + 4 * rr;
        float acc = Rs[r][c];
        for (int s = 0; s < SS; ++s) acc += HPw[r][s] * Hs[c][s];
        Scw[r][c] = acc;
        out[covBase + (size_t)t * (MM * MM) + r * MM + c] = acc;
      }
    }
    LDS_FENCE();

    // ---- 4a. Cholesky S = L Lᵀ in place (lower triangle of Scw) ----
    for (int k = 0; k < MM; ++k) {
      if (lane == k) Scw[k][k] = sqrtf(Scw[k][k]);
      LDS_FENCE();
      const float dk = Scw[k][k];
      if (lane > k && lane < MM) Scw[lane][k] = Scw[lane][k] / dk;
      LDS_FENCE();
      if (lane > k && lane < MM) {
        const float lik = Scw[lane][k];
        for (int j = k + 1; j <= lane; ++j) Scw[lane][j] -= lik * Scw[j][k];
      }
      LDS_FENCE();
    }

    // ---- 4b. Kᵀ = S⁻¹ HP : 32 RHS, lane owns column ----
    {
      float y[MM];
      for (int r = 0; r < MM; ++r) {
        float acc = HPw[r][lane];
        for (int j = 0; j < r; ++j) acc -= Scw[r][j] * y[j];
        y[r] = acc / Scw[r][r];
      }
      for (int r = MM - 1; r >= 0; --r) {
        float acc = y[r];
        for (int j = r + 1; j < MM; ++j) acc -= Scw[j][r] * y[j];
        y[r] = acc / Scw[r][r];
      }
      for (int r = 0; r < MM; ++r) KTw[r][lane] = y[r];
    }
    LDS_FENCE();

    // ---- 5. m_upd = m + Kᵀᵀ resid ----
    {
      float acc = mw[lane];
      for (int r = 0; r < MM; ++r) acc += KTw[r][lane] * rsd[r];
      muw[lane] = acc;
    }
    // ---- 6. U = P - KᵀᵀHP  (unsymmetrized; symmetrize folded into frags) ----
    for (int i = 0; i < SS; ++i) {
      float acc = Pw[i][lane];
      for (int r = 0; r < MM; ++r) acc -= KTw[r][i] * HPw[r][lane];
      Uw[i][lane] = acc;
    }
    LDS_FENCE();

    // ---- 7. T1 = F @ sym(U)   (WMMA, 32 ops) ----
    {
      v8f c00 = {}, c01 = {}, c10 = {}, c11 = {};
#pragma unroll
      for (int k4 = 0; k4 < 8; ++k4) {
        v2f b0, b1;
#pragma unroll
        for (int v = 0; v < 2; ++v) {
          const int K = 4 * k4 + 2 * v + lhi;
          b0[v] = 0.5f * (Uw[K][llo] + Uw[llo][K]);
          b1[v] = 0.5f * (Uw[K][16 + llo] + Uw[16 + llo][K]);
        }
        v2f a0 = {fa[0][k4][0], fa[0][k4][1]};
        v2f a1 = {fa[1][k4][0], fa[1][k4][1]};
        c00 = wmma4(a0, b0, c00);
        c01 = wmma4(a0, b1, c01);
        c10 = wmma4(a1, b0, c10);
        c11 = wmma4(a1, b1, c11);
      }
      // C-layout scatter of T1 into Uw (fully consumed above; same-wave DS
      // ops stay in order, so stores can't pass the loads)
#pragma unroll
      for (int v = 0; v < 8; ++v) {
        const int row = v + 8 * lhi;
        Uw[row][llo]           = c00[v];
        Uw[row][16 + llo]      = c01[v];
        Uw[16 + row][llo]      = c10[v];
        Uw[16 + row][16 + llo] = c11[v];
      }
    }
    LDS_FENCE();

    // ---- 8. P = T1 @ Fᵀ + Q   (WMMA, 32 ops; C seeded with Q frags) ----
    {
      v8f d00 = qc[0][0], d01 = qc[0][1], d10 = qc[1][0], d11 = qc[1][1];
#pragma unroll
      for (int k4 = 0; k4 < 8; ++k4) {
        v2f a0, a1;
#pragma unroll
        for (int v = 0; v < 2; ++v) {
          const int K = 4 * k4 + v + 2 * lhi;
          a0[v] = Uw[llo][K];
          a1[v] = Uw[16 + llo][K];
        }
        v2f b0 = {ftb[0][k4][0], ftb[0][k4][1]};
        v2f b1 = {ftb[1][k4][0], ftb[1][k4][1]};
        d00 = wmma4(a0, b0, d00);
        d01 = wmma4(a0, b1, d01);
        d10 = wmma4(a1, b0, d10);
        d11 = wmma4(a1, b1, d11);
      }
#pragma unroll
      for (int v = 0; v < 8; ++v) {
        const int row = v + 8 * lhi;
        Pw[row][llo]           = d00[v];
        Pw[row][16 + llo]      = d01[v];
        Pw[16 + row][llo]      = d10[v];
        Pw[16 + row][16 + llo] = d11[v];
      }
    }
    // ---- 9. m_next = F @ m_upd ----
    {
      float acc = 0.f;
      for (int k = 0; k < SS; ++k) acc += Fs[lane][k] * muw[k];
      LDS_FENCE();   // Pw stores + muw reads complete before overwriting mw
      mw[lane] = acc;
    }
    LDS_FENCE();
  }
}

extern "C" void kernel_launch(void* const* d_in, const int* in_sizes, int n_in,
                              void* d_out, int out_size, void* d_ws, size_t ws_size,
                              hipStream_t stream) {
  (void)in_sizes; (void)n_in; (void)d_ws; (void)ws_size; (void)out_size;
  const float* inp = (const float*)d_in[0];
  const float* F   = (const float*)d_in[1];
  const float* H   = (const float*)d_in[2];
  const float* Q   = (const float*)d_in[3];
  const float* R   = (const float*)d_in[4];
  const float* m0  = (const float*)d_in[5];
  const float* P0  = (const float*)d_in[6];
  float* out = (float*)d_out;

  dim3 grid(GG / WAVES);
  dim3 block(32 * WAVES);
  KalmanFilter_71124658422067_kernel<<<grid, block, 0, stream>>>(
      inp, F, H, Q, R, m0, P0, out);
}